// RNN_81011673137157
// MI455X (gfx1250) — compile-verified
//
#include <hip/hip_runtime.h>
#include <stdint.h>

// ---------------------------------------------------------------------------
// Fused RNN: out[b,c,n,t] = scan_t tanh(W1@x_t + b1 + W2@h + b2)
// CDNA5 (gfx1250) wave32, V_WMMA_F32_16X16X4_F32 + TDM async tensor loads.
// ---------------------------------------------------------------------------

#define BDIM   16
#define CDIM   64
#define NDIM   1024
#define TDIM   64
#define NT     16                 // node-column tile per workgroup
#define TC     8                  // time chunk staged per TDM transfer
#define NCHUNK (TDIM / TC)        // 8
#define XS     (TC + 1)           // padded LDS row stride (floats) for x stage (TDM pad)
#define HS     17                 // padded row stride for hidden ping-pong buffers
#define OS     12                 // padded row stride for out stage (48B -> float4 aligned)

typedef float        v2f   __attribute__((ext_vector_type(2)));
typedef float        v8f   __attribute__((ext_vector_type(8)));
typedef unsigned int u32x4 __attribute__((ext_vector_type(4)));
typedef unsigned int u32x8 __attribute__((ext_vector_type(8)));

// Only check device-side: __has_builtin(...) for AMDGPU builtins is false in
// the host (x86) pass of HIP compilation.
#if defined(__HIP_DEVICE_COMPILE__) && !__has_builtin(__builtin_amdgcn_wmma_f32_16x16x4_f32)
#error "missing __builtin_amdgcn_wmma_f32_16x16x4_f32 for gfx1250"
#endif

__device__ __forceinline__ v8f wmma4(v2f a, v2f b, v8f c) {
  // D = A(16x4) * B(4x16) + C(16x16), fp32
  return __builtin_amdgcn_wmma_f32_16x16x4_f32(false, a, false, b, (short)0, c,
                                               false, false);
}

__device__ __forceinline__ float fast_tanh(float x) {
#if defined(__HIP_DEVICE_COMPILE__) && __has_builtin(__builtin_amdgcn_tanhf)
  return __builtin_amdgcn_tanhf(x);
#else
  float r;
  asm("v_tanh_f32 %0, %1" : "=v"(r) : "v"(x));
  return r;
#endif
}

// Issue one TDM load of an x tile: tile dims (t=TC, n=NT, c=CDIM), elem=4B,
// LDS pad 1 DWORD per 8 DWORDs -> LDS layout [c][n][XS] with XS = TC+1.
__device__ __forceinline__ void tdm_load_x(const float* gsrc, uint32_t lds_off) {
  uint64_t ga = (uint64_t)(uintptr_t)gsrc;

  u32x4 g0;
  g0[0] = 1u;                                              // count=1, user desc
  g0[1] = lds_off;                                         // lds_addr (bytes)
  g0[2] = (uint32_t)ga;                                    // global_addr[31:0]
  g0[3] = ((uint32_t)(ga >> 32) & 0x01FFFFFFu) | (2u << 30); // addr[56:32] | type=2

  u32x8 g1;
  g1[0] = (2u << 16)        // data_size = 4B
        | (1u << 20)        // pad_enable
        | (2u << 22)        // pad_interval: every 8 DWORDs
        | (0u << 25);       // pad_amount: 1 DWORD
  g1[1] = ((uint32_t)TDIM & 0xFFFFu) << 16;                        // tensor_dim0 lo16
  g1[2] = ((uint32_t)TDIM >> 16) | (((uint32_t)NT & 0xFFFFu) << 16); // dim0 hi | dim1 lo
  g1[3] = ((uint32_t)NT >> 16) | ((uint32_t)TC << 16);             // dim1 hi | tile_dim0
  g1[4] = (uint32_t)NT | ((uint32_t)CDIM << 16);                   // tile_dim1 | tile_dim2
  g1[5] = (uint32_t)TDIM;                                          // dim0_stride lo32 (=T)
  g1[6] = 0u | (((uint32_t)(NDIM * TDIM) & 0xFFFFu) << 16);        // dim0_stride hi | dim1_stride lo
  g1[7] = (uint32_t)(NDIM * TDIM) >> 16;                           // dim1_stride[47:16]

  u32x4 g2 = { (uint32_t)CDIM, 0u, 0u, 0u };               // tensor_dim2
  u32x4 g3 = { 0u, 0u, 0u, 0u };

  asm volatile("tensor_load_to_lds %0, %1, %2, %3"
               :: "s"(g0), "s"(g1), "s"(g2), "s"(g3)
               : "memory");
}

__global__ __launch_bounds__(128)
void RNN_81011673137157_kernel(const float* __restrict__ x,
                               const float* __restrict__ w1,
                               const float* __restrict__ b1,
                               const float* __restrict__ w2,
                               const float* __restrict__ b2,
                               float* __restrict__ out) {
  __shared__ __align__(16) float xs[2][CDIM * NT * XS]; // 73728 B (TDM targets)
  __shared__ __align__(16) float hb[2][CDIM * HS];      //  8704 B (hidden ping-pong)
  __shared__ __align__(16) float os[CDIM * NT * OS];    // 49152 B (output stage)

  const int tid   = threadIdx.x;
  const int lane  = tid & 31;
  const int wv    = tid >> 5;        // 0..3 : M-tile (c_out rows wv*16..wv*16+15)
  const int nn    = lane & 15;       // column (node) within tile / M row for A
  const int hi    = lane >> 4;       // 0/1 lane-half
  const int khalf = hi * 2;          // K sub-offset per A/B layout

  const int bb = blockIdx.x >> 6;          // batch index
  const int n0 = (blockIdx.x & 63) * NT;   // node tile base

  const float* xbase = x + ((size_t)bb * CDIM * NDIM + n0) * TDIM;

  // ---- A-operand weight tiles (registers, layout: lane=(M=lane&15),
  // VGPR v holds K = 4*kk + khalf + v) -------------------------------------
  const int mrow = wv * 16 + nn;
  v2f A1[16], A2[16];
#pragma unroll
  for (int kk = 0; kk < 16; ++kk) {
    int k0 = kk * 4 + khalf;
    A1[kk] = *(const v2f*)(w1 + mrow * CDIM + k0);
    A2[kk] = *(const v2f*)(w2 + mrow * CDIM + k0);
  }
  // Bias per accumulator element: D vgpr v holds row M = v + 8*hi.
  float bias[8];
#pragma unroll
  for (int v = 0; v < 8; ++v) {
    int m = wv * 16 + v + hi * 8;
    bias[v] = b1[m] + b2[m];
  }

  // h(0) = 0
  for (int i = tid; i < CDIM * HS; i += 128) hb[0][i] = 0.f;

  // Prologue: prefetch chunks 0 and 1 via TDM (wave 0 only; EXEC ignored).
  if (wv == 0) {
    tdm_load_x(xbase + 0 * TC, (uint32_t)(uintptr_t)&xs[0][0]);
    tdm_load_x(xbase + 1 * TC, (uint32_t)(uintptr_t)&xs[1][0]);
  }

  int p = 0;
  for (int c = 0; c < NCHUNK; ++c) {
    if (wv == 0) {
      if (c < NCHUNK - 1) __builtin_amdgcn_s_wait_tensorcnt(1);
      else                __builtin_amdgcn_s_wait_tensorcnt(0);
    }
    __syncthreads();   // x chunk ready (and h0 zeroed on first pass)

    const float* xc = xs[c & 1];
    const int t0 = c * TC;

    for (int tl = 0; tl < TC; ++tl) {
      const float* hp = hb[p];
      float*       hn = hb[p ^ 1];

      v8f acc;
#pragma unroll
      for (int v = 0; v < 8; ++v) acc[v] = bias[v];

      // acc += W1 @ x_t   (B operand: K=c_in from LDS x stage, fixed t)
#pragma unroll
      for (int kk = 0; kk < 16; ++kk) {
        int c0 = kk * 4 + khalf;
        v2f bx;
        bx[0] = xc[(c0 * NT + nn) * XS + tl];
        bx[1] = xc[((c0 + 1) * NT + nn) * XS + tl];
        acc = wmma4(A1[kk], bx, acc);
      }
      // acc += W2 @ h     (B operand: K=c_in from hidden buffer)
#pragma unroll
      for (int kk = 0; kk < 16; ++kk) {
        int c0 = kk * 4 + khalf;
        v2f bh;
        bh[0] = hp[c0 * HS + nn];
        bh[1] = hp[(c0 + 1) * HS + nn];
        acc = wmma4(A2[kk], bh, acc);
      }

#pragma unroll
      for (int v = 0; v < 8; ++v) {
        float h = fast_tanh(acc[v]);
        int m = wv * 16 + v + hi * 8;
        hn[m * HS + nn] = h;                  // next-step B operand
        os[(m * NT + nn) * OS + tl] = h;      // output staging
      }
      p ^= 1;
      __syncthreads();
    }

    // Prefetch chunk c+2 into the buffer whose reads just completed.
    if (wv == 0 && c + 2 < NCHUNK) {
      tdm_load_x(xbase + (c + 2) * TC, (uint32_t)(uintptr_t)&xs[c & 1][0]);
    }

    // Flush this chunk: out[b, m, n0+n, t0 .. t0+TC-1] as contiguous float4s.
    for (int r = tid; r < CDIM * NT; r += 128) {
      int m = r >> 4, n = r & 15;
      const float4* src = (const float4*)&os[r * OS];
      float4 lo = src[0];
      float4 hi4 = src[1];
      float* dst = out + (((size_t)bb * CDIM + m) * NDIM + (n0 + n)) * TDIM + t0;
      *(float4*)(dst)     = lo;
      *(float4*)(dst + 4) = hi4;
    }
    __syncthreads();   // os reusable next chunk
  }
}

extern "C" void kernel_launch(void* const* d_in, const int* in_sizes, int n_in,
                              void* d_out, int out_size, void* d_ws, size_t ws_size,
                              hipStream_t stream) {
  (void)in_sizes; (void)n_in; (void)d_ws; (void)ws_size; (void)out_size;
  const float* x  = (const float*)d_in[0];
  const float* w1 = (const float*)d_in[1];
  const float* b1 = (const float*)d_in[2];
  const float* w2 = (const float*)d_in[3];
  const float* b2 = (const float*)d_in[4];
  float* out = (float*)d_out;

  dim3 grid(BDIM * (NDIM / NT));   // 1024 workgroups
  dim3 block(128);                 // 4 waves (wave32)
  RNN_81011673137157_kernel<<<grid, block, 0, stream>>>(x, w1, b1, w2, b2, out);
}